// PNANet_19688130085213
// MI455X (gfx1250) — compile-verified
//
#include <hip/hip_runtime.h>
#include <math.h>

#define N_NODES 25000
#define N_EDGES 400000
#define ROW_TILES ((N_NODES + 15) / 16)   // 1563

typedef float v2f __attribute__((ext_vector_type(2)));
typedef float v8f __attribute__((ext_vector_type(8)));

__device__ __forceinline__ v8f wmma_f32_k4(v2f a, v2f b, v8f c) {
  // D = A(16x4,f32) * B(4x16,f32) + C(16x16,f32)
  return __builtin_amdgcn_wmma_f32_16x16x4_f32(false, a, false, b, (short)0, c,
                                               false, false);
}

// ---------------------------------------------------------------- utilities
__global__ void k_fill(float* __restrict__ p, float v, long n) {
  long i = (long)blockIdx.x * blockDim.x + threadIdx.x;
  if (i < n) p[i] = v;
}

__global__ void k_degree(const int* __restrict__ dst, float* __restrict__ cnt,
                         int e) {
  int i = blockIdx.x * blockDim.x + threadIdx.x;
  if (i < e) atomicAdd(&cnt[dst[i]], 1.0f);
}

__global__ void k_avglog(const float* __restrict__ cnt,
                         float* __restrict__ stats, int n) {
  int i = blockIdx.x * blockDim.x + threadIdx.x;
  if (i < n) atomicAdd(&stats[0], logf(cnt[i] + 1.0f));
}

// ------------------------------------------------------- generic WMMA GEMM
// C[m, c0+col] = sum_k A[m,k] * B[c0+col, k] (+bias) (+relu)
// A: nrows x lda (uses first K cols), B: row-major weight[out][in] stride ldb.
// one wave computes a 16x16 tile; col tile = blockIdx.y*wavesPerBlock + wave.
__global__ void k_gemm(const float* __restrict__ A, int lda,
                       const float* __restrict__ B, int ldb,
                       const float* __restrict__ bias, float* __restrict__ C,
                       int ldc, int nrows, int K, int act) {
  int lane = threadIdx.x & 31;
  int wave = threadIdx.x >> 5;
  int ct = blockIdx.y * (blockDim.x >> 5) + wave;
  int c0 = ct * 16;
  int half = lane >> 4;
  int rc = lane & 15;  // A row within tile / B-D column

  long ar = (long)blockIdx.x * 16 + rc;
  if (ar >= nrows) ar = nrows - 1;  // clamp loads; stores are masked
  const float* Arow = A + ar * (long)lda;
  const float* Brow = B + (long)(c0 + rc) * ldb;

  v8f acc = {};
  for (int kb = 0; kb < K; kb += 4) {
    int k = kb + 2 * half;
    v2f a = {Arow[k], Arow[k + 1]};
    v2f b = {Brow[k], Brow[k + 1]};
    acc = wmma_f32_k4(a, b, acc);
  }

  float bv = bias ? bias[c0 + rc] : 0.0f;
  long n0 = (long)blockIdx.x * 16;
  for (int r = 0; r < 8; ++r) {
    long m = n0 + r + 8 * half;
    if (m < nrows) {
      float v = acc[r] + bv;
      if (act) v = fmaxf(v, 0.0f);
      C[m * (long)ldc + c0 + rc] = v;
    }
  }
}

// ------------------------------------------------------------- aggregation
// Base layout: (N, T=4, slot=4, 64) -> slot 0=sum 1=min 2=max 3=sumsq
__global__ void k_init_base(float* __restrict__ Base, long n) {
  long i = (long)blockIdx.x * blockDim.x + threadIdx.x;
  if (i < n) {
    int slot = (int)((i >> 6) & 3);
    Base[i] = (slot == 1) ? __FLT_MAX__ : (slot == 2) ? -__FLT_MAX__ : 0.0f;
  }
}

__global__ void k_edges(const float* __restrict__ NI,
                        const float* __restrict__ NJ,
                        const float* __restrict__ pre_b,
                        const int* __restrict__ src,
                        const int* __restrict__ dst,
                        float* __restrict__ Base) {
  int e = blockIdx.x;
  int c = threadIdx.x;  // 0..255  (= t*64+g)
  int s = src[e], d = dst[e];
  float h = NI[(long)d * 256 + c] + NJ[(long)s * 256 + c] + pre_b[c];
  int t = c >> 6, g = c & 63;
  float* B = Base + (long)d * 1024 + t * 256;
  atomicAdd(B + g, h);
  atomicMin(B + 64 + g, h);
  atomicMax(B + 128 + g, h);
  atomicAdd(B + 192 + g, h * h);
}

__global__ void k_finalize(float* __restrict__ Base,
                           const float* __restrict__ cnt, long total) {
  long i = (long)blockIdx.x * blockDim.x + threadIdx.x;  // over N*256
  if (i >= total) return;
  long n = i >> 8;
  int c = (int)(i & 255);
  int t = c >> 6, g = c & 63;
  float* p = Base + n * 1024 + t * 256;
  float cn = cnt[n];
  float dg = fmaxf(cn, 1.0f);
  float s = p[g], mn = p[64 + g], mx = p[128 + g], s2 = p[192 + g];
  float mean = s / dg;
  float var = fmaxf(s2 / dg - mean * mean, 0.0f);
  float sd = sqrtf(var + 1e-5f);
  if (cn <= 0.0f) { mn = 0.0f; mx = 0.0f; }
  p[g] = mean; p[64 + g] = mn; p[128 + g] = mx; p[192 + g] = sd;
}

// ----------------------------------------------------------------- post op
// P[n, t*16+g] = X*Wx + amp*(base*W1) + (1/amp)*(base*W2) + base*W3 + b
// Wp: (T,16,832) with f slices [0,64)=x, [64,320), [320,576), [576,832).
__global__ void k_post(const float* __restrict__ Base,
                       const float* __restrict__ X,
                       const float* __restrict__ Wp,
                       const float* __restrict__ bp,
                       const float* __restrict__ cnt,
                       const float* __restrict__ stats,
                       float* __restrict__ P, int nrows) {
  int lane = threadIdx.x & 31;
  int t = threadIdx.x >> 5;  // wave = tower (blockDim=128)
  int half = lane >> 4;
  int rc = lane & 15;

  long ar = (long)blockIdx.x * 16 + rc;
  if (ar >= nrows) ar = nrows - 1;
  const float* Ax = X + ar * 64;
  const float* Ab = Base + ar * 1024 + t * 256;
  const float* Bw = Wp + (long)t * 16 * 832 + (long)rc * 832;

  v8f cx = {}, c1 = {}, c2 = {}, c3 = {};
  for (int kb = 0; kb < 64; kb += 4) {
    int k = kb + 2 * half;
    v2f a = {Ax[k], Ax[k + 1]};
    v2f b = {Bw[k], Bw[k + 1]};
    cx = wmma_f32_k4(a, b, cx);
  }
  for (int kb = 0; kb < 256; kb += 4) {
    int k = kb + 2 * half;
    v2f a = {Ab[k], Ab[k + 1]};
    v2f b1 = {Bw[64 + k], Bw[64 + k + 1]};
    c1 = wmma_f32_k4(a, b1, c1);
    v2f b2 = {Bw[320 + k], Bw[320 + k + 1]};
    c2 = wmma_f32_k4(a, b2, c2);
    v2f b3 = {Bw[576 + k], Bw[576 + k + 1]};
    c3 = wmma_f32_k4(a, b3, c3);
  }

  float avg_log = stats[0] / (float)nrows;
  float bias = bp[t * 16 + rc];
  long n0 = (long)blockIdx.x * 16;
  for (int r = 0; r < 8; ++r) {
    long m = n0 + r + 8 * half;
    if (m < nrows) {
      float dg = fmaxf(cnt[m], 1.0f);
      float amp = logf(dg + 1.0f) / avg_log;
      float v = cx[r] + amp * c1[r] + c2[r] / amp + c3[r] + bias;
      P[m * 64 + t * 16 + rc] = v;
    }
  }
}

// ------------------------------------------------------------------- BN
__global__ void k_bn_stats(const float* __restrict__ H, float* __restrict__ sum,
                           float* __restrict__ sumsq, int n) {
  int c = threadIdx.x;  // 64 channels
  float s = 0.0f, s2 = 0.0f;
  for (int i = blockIdx.x; i < n; i += gridDim.x) {
    float v = H[(long)i * 64 + c];
    s += v;
    s2 += v * v;
  }
  atomicAdd(&sum[c], s);
  atomicAdd(&sumsq[c], s2);
}

__global__ void k_bn_apply(float* __restrict__ H,
                           const float* __restrict__ sum,
                           const float* __restrict__ sumsq,
                           const float* __restrict__ g,
                           const float* __restrict__ b, int n) {
  long i = (long)blockIdx.x * blockDim.x + threadIdx.x;
  if (i >= (long)n * 64) return;
  int c = (int)(i & 63);
  float m = sum[c] / (float)n;
  float var = fmaxf(sumsq[c] / (float)n - m * m, 0.0f);
  float v = H[i];
  v = g[c] * (v - m) * rsqrtf(var + 1e-5f) + b[c];
  H[i] = fmaxf(v, 0.0f);
}

// ================================================================= launch
extern "C" void kernel_launch(void* const* d_in, const int* in_sizes, int n_in,
                              void* d_out, int out_size, void* d_ws,
                              size_t ws_size, hipStream_t stream) {
  const int N = N_NODES, E = N_EDGES;
  const float* x = (const float*)d_in[0];
  const int* src = (const int*)d_in[1];
  const int* dst = src + E;

  const float* pre_w[2] = {(const float*)d_in[2], (const float*)d_in[8]};
  const float* pre_b[2] = {(const float*)d_in[3], (const float*)d_in[9]};
  const float* post_w[2] = {(const float*)d_in[4], (const float*)d_in[10]};
  const float* post_b[2] = {(const float*)d_in[5], (const float*)d_in[11]};
  const float* lin_w[2] = {(const float*)d_in[6], (const float*)d_in[12]};
  const float* lin_b[2] = {(const float*)d_in[7], (const float*)d_in[13]};
  const float* bn_g[2] = {(const float*)d_in[14], (const float*)d_in[16]};
  const float* bn_b[2] = {(const float*)d_in[15], (const float*)d_in[17]};
  const float* mlp_w1 = (const float*)d_in[18];
  const float* mlp_b1 = (const float*)d_in[19];
  const float* mlp_w2 = (const float*)d_in[20];
  const float* mlp_b2 = (const float*)d_in[21];

  float* ws = (float*)d_ws;
  float* cnt = ws;                       // N
  float* stats = cnt + N;                // 256: [0]=sum log, [16..79] bn sum,
                                         //      [80..143] bn sumsq
  float* NI = stats + 256;               // N*256
  float* NJ = NI + (size_t)N * 256;      // N*256
  float* Base = NJ + (size_t)N * 256;    // N*1024
  float* P = Base + (size_t)N * 1024;    // N*64
  float* H = P + (size_t)N * 64;         // N*64

  // degree / avg_log
  k_fill<<<(N + 255) / 256, 256, 0, stream>>>(cnt, 0.0f, N);
  k_fill<<<1, 256, 0, stream>>>(stats, 0.0f, 256);
  k_degree<<<(E + 255) / 256, 256, 0, stream>>>(dst, cnt, E);
  k_avglog<<<(N + 255) / 256, 256, 0, stream>>>(cnt, stats, N);

  for (int L = 0; L < 2; ++L) {
    const float* X = (L == 0) ? x : H;
    // ni / nj:  (N x 64) @ wi[t].T -> (N x 256), wi row c -> pre_w[c*128 + f]
    k_gemm<<<dim3(ROW_TILES, 2), 256, 0, stream>>>(X, 64, pre_w[L], 128,
                                                   nullptr, NI, 256, N, 64, 0);
    k_gemm<<<dim3(ROW_TILES, 2), 256, 0, stream>>>(X, 64, pre_w[L] + 64, 128,
                                                   nullptr, NJ, 256, N, 64, 0);
    // segment reductions
    k_init_base<<<(int)(((long)N * 1024 + 255) / 256), 256, 0, stream>>>(
        Base, (long)N * 1024);
    k_edges<<<E, 256, 0, stream>>>(NI, NJ, pre_b[L], src, dst, Base);
    k_finalize<<<(int)(((long)N * 256 + 255) / 256), 256, 0, stream>>>(
        Base, cnt, (long)N * 256);
    // post towers (WMMA) with amp folded outside the GEMM
    k_post<<<ROW_TILES, 128, 0, stream>>>(Base, X, post_w[L], post_b[L], cnt,
                                          stats, P, N);
    // lin: (N x 64) @ lin_w.T + lin_b -> H
    k_gemm<<<dim3(ROW_TILES, 1), 128, 0, stream>>>(P, 64, lin_w[L], 64,
                                                   lin_b[L], H, 64, N, 64, 0);
    // BN + ReLU (in place on H)
    k_fill<<<1, 256, 0, stream>>>(stats + 16, 0.0f, 128);
    k_bn_stats<<<256, 64, 0, stream>>>(H, stats + 16, stats + 80, N);
    k_bn_apply<<<(int)(((long)N * 64 + 255) / 256), 256, 0, stream>>>(
        H, stats + 16, stats + 80, bn_g[L], bn_b[L], N);
  }

  // MLP: relu(H @ w1.T + b1) -> P ; P @ w2.T + b2 -> out (N x 32)
  k_gemm<<<dim3(ROW_TILES, 1), 128, 0, stream>>>(H, 64, mlp_w1, 64, mlp_b1, P,
                                                 64, N, 64, 1);
  k_gemm<<<dim3(ROW_TILES, 1), 64, 0, stream>>>(P, 64, mlp_w2, 64, mlp_b2,
                                                (float*)d_out, 32, N, 64, 0);
}